// NodeProcessor_17386027614329
// MI455X (gfx1250) — compile-verified
//
#include <hip/hip_runtime.h>

typedef float v2f __attribute__((ext_vector_type(2)));
typedef float v8f __attribute__((ext_vector_type(8)));

#define FN 128   // node features
#define FE 48    // edge features
#define FG 64    // global features
#define FO 128   // output features
// K layout of the implicit concat: [0,128)=nodes, [128,176)=agg, [176,240)=globals

// ---------------------------------------------------------------------------
// Pass 0: zero the segment-sum accumulator in workspace (must happen per call)
// ---------------------------------------------------------------------------
__global__ void zero_f32(float* __restrict__ p, long long n) {
  long long i = (long long)blockIdx.x * blockDim.x + threadIdx.x;
  long long s = (long long)gridDim.x * blockDim.x;
  for (; i < n; i += s) p[i] = 0.0f;
}

// ---------------------------------------------------------------------------
// Pass 1: segment_sum(edges, receivers) -> agg[N_NODES, FE]
// One thread per (edge, feature) element; consecutive threads share an edge,
// so edge reads are fully coalesced. Edges are read-once (307 MB) -> NT loads
// so they don't evict the hot agg buffer (19 MB) from the 192 MB L2.
// Guaranteed-native fire-and-forget atomic via inline global_atomic_add_f32
// (no-return form; drained by S_ENDPGM's implicit wait-idle).
// ---------------------------------------------------------------------------
__global__ void scatter_edges(const float* __restrict__ edges,
                              const long long* __restrict__ recv,
                              float* __restrict__ agg,
                              long long n_elems) {
  long long i = (long long)blockIdx.x * blockDim.x + threadIdx.x;
  long long s = (long long)gridDim.x * blockDim.x;
  for (; i < n_elems; i += s) {
    long long e = i / FE;                    // magic-multiply, cheap
    int f = (int)(i - e * FE);
    long long r = recv[e];                   // 48 lanes share -> cache hit
    float v = __builtin_nontemporal_load(&edges[i]);
    float* dst = &agg[r * FE + f];
    asm volatile("global_atomic_add_f32 %0, %1, off"
                 :
                 : "v"(dst), "v"(v)
                 : "memory");
  }
}

// ---------------------------------------------------------------------------
// Pass 2: fused gather + concat + GEMM + bias + ReLU using fp32 WMMA.
// One wave -> one 16x128 output tile. 60 K-chunks of 4 (fp32 WMMA K=4),
// 8 N-tiles of 16 -> 480 v_wmma_f32_16x16x4_f32 per tile.
//
// A-fragment layout (16x4 f32): lane<16 holds {K=k0,k0+1}, lane>=16 holds
// {K=k0+2,k0+3} of row (lane&15). B-fragment mirrors with N=(lane&15).
// ---------------------------------------------------------------------------
#define DO_KCHUNK(AFRAG, KBASE)                                                \
  {                                                                            \
    v2f afrag = (AFRAG);                                                       \
    _Pragma("unroll")                                                          \
    for (int j = 0; j < 8; ++j) {                                              \
      const int col = j * 16 + row;                                            \
      v2f bfrag;                                                               \
      bfrag.x = W[(size_t)((KBASE) + 2 * hi)     * FO + col];                  \
      bfrag.y = W[(size_t)((KBASE) + 2 * hi + 1) * FO + col];                  \
      acc[j] = __builtin_amdgcn_wmma_f32_16x16x4_f32(                          \
          false, afrag, false, bfrag, (short)0, acc[j], false, false);         \
    }                                                                          \
  }

__global__ void __launch_bounds__(128)
gnn_node_mlp(const float* __restrict__ nodes,
             const float* __restrict__ agg,
             const float* __restrict__ globs,
             const long long* __restrict__ nid,
             const float* __restrict__ W,
             const float* __restrict__ bias,
             float* __restrict__ out,
             int n_nodes) {
  const int lane = threadIdx.x & 31;
  const int wave = threadIdx.x >> 5;
  const int tile = blockIdx.x * 4 + wave;
  const int n_tiles = (n_nodes + 15) >> 4;
  if (tile >= n_tiles) return;  // wave-uniform: EXEC stays all-1s inside

  const int row = lane & 15;
  const int hi  = lane >> 4;  // 0 or 1 (which K-pair / which M-half we own)
  int node = tile * 16 + row;
  if (node >= n_nodes) node = n_nodes - 1;  // clamp (keeps EXEC full for WMMA)

  const float* xn = nodes + (size_t)node * FN;
  const float* xa = agg   + (size_t)node * FE;
  const float* xg = globs + (size_t)nid[node] * FG;

  v8f acc[8];
#pragma unroll
  for (int j = 0; j < 8; ++j) acc[j] = (v8f){0.f,0.f,0.f,0.f,0.f,0.f,0.f,0.f};

  // K region 1: node features (read-once stream -> NT loads), chunks 0..31
  for (int c = 0; c < FN / 4; ++c)
    DO_KCHUNK(__builtin_nontemporal_load((const v2f*)(xn + 4 * c + 2 * hi)),
              4 * c);
  // K region 2: aggregated edge features (L2-hot), chunks 32..43
  for (int c = 0; c < FE / 4; ++c)
    DO_KCHUNK(*(const v2f*)(xa + 4 * c + 2 * hi), FN + 4 * c);
  // K region 3: broadcast globals (tiny, cached), chunks 44..59
  for (int c = 0; c < FG / 4; ++c)
    DO_KCHUNK(*(const v2f*)(xg + 4 * c + 2 * hi), FN + FE + 4 * c);

  // C/D layout: VGPR r -> M = r + 8*hi, N = (lane&15) within each 16-col tile
  float* otile = out + (size_t)tile * 16 * FO;
#pragma unroll
  for (int j = 0; j < 8; ++j) {
    const int col = j * 16 + row;
    const float bb = bias[col];
#pragma unroll
    for (int r = 0; r < 8; ++r) {
      const int m = r + 8 * hi;
      const int orow = tile * 16 + m;
      if (orow < n_nodes) {
        float v = acc[j][r] + bb;
        v = v > 0.f ? v : 0.f;
        __builtin_nontemporal_store(v, &otile[(size_t)m * FO + col]);
      }
    }
  }
}

// ---------------------------------------------------------------------------
extern "C" void kernel_launch(void* const* d_in, const int* in_sizes, int n_in,
                              void* d_out, int out_size, void* d_ws, size_t ws_size,
                              hipStream_t stream) {
  const float*     nodes = (const float*)d_in[0];
  const float*     edges = (const float*)d_in[1];
  const float*     globs = (const float*)d_in[2];
  const float*     W     = (const float*)d_in[3];
  const float*     bias  = (const float*)d_in[4];
  const long long* recv  = (const long long*)d_in[5];  // int64 in reference
  const long long* nid   = (const long long*)d_in[6];  // int64 in reference
  float* out = (float*)d_out;

  const int       n_nodes = in_sizes[0] / FN;
  const long long n_edges = (long long)in_sizes[1] / FE;

  float* agg = (float*)d_ws;  // n_nodes * FE floats (~19.2 MB)
  const long long agg_elems = (long long)n_nodes * FE;

  zero_f32<<<1024, 256, 0, stream>>>(agg, agg_elems);

  const long long scat_elems = n_edges * FE;  // 76.8M
  scatter_edges<<<8192, 256, 0, stream>>>(edges, recv, agg, scat_elems);

  const int n_tiles = (n_nodes + 15) / 16;        // 6250
  const int blocks  = (n_tiles + 3) / 4;          // 4 waves per block
  gnn_node_mlp<<<blocks, 128, 0, stream>>>(nodes, agg, globs, nid, W, bias,
                                           out, n_nodes);
}